// BatchedProblemWrapper_15590731285215
// MI455X (gfx1250) — compile-verified
//
#include <hip/hip_runtime.h>
#include <hip/hip_bf16.h>

// Problem constants from the reference
#define BB 8
#define DD 1024
#define GG 32
#define KK 64
#define NPP 1024

typedef __attribute__((ext_vector_type(2))) float v2f;
typedef __attribute__((ext_vector_type(8))) float v8f;

// One wave (32 threads) per (batch b, 16-row tile of NP).
// grid = (NP/16, B), block = 32.
__global__ void __launch_bounds__(32)
bbob_group_elliptic_kernel(const float* __restrict__ x,          // (B,NP,D)
                           const float* __restrict__ weights,    // (B,G)
                           const float* __restrict__ xopt,       // (B,D)
                           const float* __restrict__ R,          // (K,K) row-major
                           const int*   __restrict__ gidx,       // (B,G,K)
                           const unsigned char* __restrict__ vmask, // (B,G,K) bool
                           const int*   __restrict__ gcounts,    // (B,)
                           float* __restrict__ out)              // (B,NP)
{
    // R staged interleaved by row pairs so one aligned ds_load_b64 yields a
    // complete B fragment {R[k0][col], R[k0+1][col]} in a VGPR pair:
    //   sRi[(k>>1)*128 + col*2 + (k&1)] = R[k][col]
    __shared__ float sRi[KK * KK];  // 16 KB
    __shared__ float sCoeff[KK];    // elliptic coeffs (group-invariant)
    __shared__ int   sIdx[KK];
    __shared__ float sXop[KK];      // xopt gathered at group indices
    __shared__ float sMc[KK];       // w * (valid ? coeff : 0)

    const int lane = threadIdx.x;        // 0..31
    const int half = lane >> 4;          // 0 or 1
    const int l16  = lane & 15;
    const int b    = blockIdx.y;
    const int row0 = blockIdx.x * 16;    // tile of 16 NP rows

    // Stage R (interleaved) and the coefficient table once per block
    for (int i = lane; i < KK * KK; i += 32) {
        const int k   = i >> 6;
        const int col = i & 63;
        sRi[(k >> 1) * 128 + col * 2 + (k & 1)] = R[i];
    }
    for (int k = lane; k < KK; k += 32) {
        // coeffs = 1e6 ** (k/(K-1))  -> exp(ln(1e6) * k/63)
        sCoeff[k] = __expf(13.815510557964274f * ((float)k * (1.0f / 63.0f)));
    }
    __syncthreads();

    const int gcount = gcounts[b];

    // Per-lane partial accumulators: pacc[nt][v] = sum over groups of
    // mc*w*z_rot^2 for row (half*8+v), column (nt*16+l16). Cross-lane
    // reduction over the 16 columns of each half is deferred to the end.
    float pacc[4][8];
#pragma unroll
    for (int nt = 0; nt < 4; ++nt)
#pragma unroll
        for (int v = 0; v < 8; ++v) pacc[nt][v] = 0.0f;

    // Row pointer for the A-tile gather: lane&15 selects the M row.
    const float* xrow = x + ((size_t)b * NPP + (size_t)(row0 + l16)) * DD;

    for (int g = 0; g < GG; ++g) {
        if (g >= gcount) break;  // group_valid is a contiguous prefix
        const float w = weights[b * GG + g];

        // Stage per-group indices / gathered xopt / masked weighted coeffs
        __syncthreads();
        for (int k = lane; k < KK; k += 32) {
            const int idx = gidx[((size_t)b * GG + g) * KK + k];
            sIdx[k] = idx;
            sXop[k] = xopt[b * DD + idx];
            sMc[k]  = vmask[((size_t)b * GG + g) * KK + k] ? (w * sCoeff[k]) : 0.0f;
        }
        __syncthreads();

        // Gather the full 16x64 A-tile (z_sub) into 16 v2f fragments.
        // f32 WMMA A layout (16x4): lane&15 = M, K = 2*half + v within a chunk.
        v2f A[16];
#pragma unroll
        for (int kc = 0; kc < 16; ++kc) {
            const int k0 = kc * 4 + 2 * half;
            const int i0 = sIdx[k0];
            const int i1 = sIdx[k0 + 1];
            v2f a;
            a.x = xrow[i0] - sXop[k0];
            a.y = xrow[i1] - sXop[k0 + 1];
            A[kc] = a;
        }

        // 4 N-tiles of the 16x64 rotated output
#pragma unroll
        for (int nt = 0; nt < 4; ++nt) {
            // Lane-dependent base into the interleaved R: column nt*16+l16,
            // row-pair selected by half; kc steps by 256 floats (1024 B).
            const float* bp = &sRi[half * 128 + (nt * 16 + l16) * 2];
            v8f acc = {};
#pragma unroll
            for (int kc = 0; kc < 16; ++kc) {
                // One aligned 8-byte LDS load -> {R[k0][col], R[k0+1][col]}
                const v2f Bf = *(const v2f*)(bp + kc * 256);
                acc = __builtin_amdgcn_wmma_f32_16x16x4_f32(
                    /*neg_a=*/false, A[kc],
                    /*neg_b=*/false, Bf,
                    /*c_mod=*/(short)0, acc,
                    /*reuse_a=*/false, /*reuse_b=*/false);
            }

            // Elliptic masked partial: lane owns column (nt*16 + l16).
            const float mcw = sMc[nt * 16 + l16];
#pragma unroll
            for (int v = 0; v < 8; ++v) {
                pacc[nt][v] += mcw * acc[v] * acc[v];
            }
        }
    }

    // Final reduction: sum the 4 N-tile partials, then reduce over the 16
    // lanes of each half (the 16 columns). Done once, not per group.
    if (true) {
#pragma unroll
        for (int v = 0; v < 8; ++v) {
            float s = (pacc[0][v] + pacc[1][v]) + (pacc[2][v] + pacc[3][v]);
            s += __shfl_xor(s, 1, 32);
            s += __shfl_xor(s, 2, 32);
            s += __shfl_xor(s, 4, 32);
            s += __shfl_xor(s, 8, 32);
            pacc[0][v] = s;  // replicated across each 16-lane half
        }
    }

    // Store: lanes 0..15 hold rows 0..7 (replicated), lanes 16..31 rows 8..15.
    if (l16 == 0) {
#pragma unroll
        for (int v = 0; v < 8; ++v) {
            out[(size_t)b * NPP + row0 + half * 8 + v] = pacc[0][v];
        }
    }
}

extern "C" void kernel_launch(void* const* d_in, const int* in_sizes, int n_in,
                              void* d_out, int out_size, void* d_ws, size_t ws_size,
                              hipStream_t stream) {
    (void)in_sizes; (void)n_in; (void)d_ws; (void)ws_size; (void)out_size;
    const float*         x       = (const float*)d_in[0];
    const float*         weights = (const float*)d_in[1];
    const float*         xopt    = (const float*)d_in[2];
    const float*         R       = (const float*)d_in[3];
    const int*           gidx    = (const int*)d_in[4];
    const unsigned char* vmask   = (const unsigned char*)d_in[5];  // bool -> 1 byte
    const int*           gcounts = (const int*)d_in[6];
    float* out = (float*)d_out;

    dim3 grid(NPP / 16, BB);
    dim3 block(32);
    bbob_group_elliptic_kernel<<<grid, block, 0, stream>>>(
        x, weights, xopt, R, gidx, vmask, gcounts, out);
}